// CrossAttentionConditionInjection_3118146257677
// MI455X (gfx1250) — compile-verified
//
#include <hip/hip_runtime.h>
#include <hip/hip_bf16.h>

// CrossAttentionConditionInjection, algebraically collapsed.
//
// K/V come from ONE condition token broadcast over all S key slots, so the
// softmax row is uniform (exactly 1/2048) and attention output == V row:
//   out[b,s,:] = ((condition[b] @ Wv^T + bv) @ Wo^T + bo)   for every s.
// hidden_states / Wq / Wk do not affect the result.
//
// Cost model (MI455X, 23.3 TB/s HBM):
//   - 2 small f32 GEMMs [2,1024]x[1024,1024]^T  -> ~8 MB weight reads
//   - broadcast store of [2,2048,1024] f32      -> 16.8 MB NT stores
//   => ~1-3 us total, store-bandwidth bound. GEMMs run on the WMMA pipe
//      (v_wmma_f32_16x16x4_f32, full fp32 to match the f32 reference).

#define D_DIM 1024
#define S_DIM 2048
#define B_DIM 2

typedef __attribute__((ext_vector_type(2))) float v2f;
typedef __attribute__((ext_vector_type(4))) float v4f;
typedef __attribute__((ext_vector_type(8))) float v8f;

// O[2, 1024] = X[2, 1024] @ W[1024,1024]^T + bias[1024]
// Grid: 64 blocks (one 16-col N tile each), 128 threads (4 waves splitting K).
__global__ __launch_bounds__(128) void smallgemm_wmma_f32(
    const float* __restrict__ X,
    const float* __restrict__ W,
    const float* __restrict__ bias,
    float* __restrict__ O) {
  const int lane = threadIdx.x & 31;
  const int wave = threadIdx.x >> 5;        // 0..3, K-chunk owner
  const int n0   = blockIdx.x << 4;         // N tile base (16 cols)
  const int row  = lane & 15;               // M index (only 0,1 are real)
  const int khalf = (lane >> 4) << 1;       // lanes 16-31 carry K+2,K+3
  const int ncol = n0 + (lane & 15);

  // Zero-pad A rows 2..15 via multiply (no EXEC divergence around WMMA).
  const float mrow = (row < B_DIM) ? 1.0f : 0.0f;
  const float* __restrict__ xrow = X + (row & 1) * D_DIM;
  const float* __restrict__ wrow = W + (size_t)ncol * D_DIM;

  v8f acc = {};
  const int kb = wave << 8;                 // 256-wide K chunk per wave
#pragma unroll 4
  for (int k0 = kb; k0 < kb + 256; k0 += 4) {
    v2f a, b;
    a.x = xrow[k0 + khalf]     * mrow;      // A[row, k0+khalf+0]
    a.y = xrow[k0 + khalf + 1] * mrow;      // A[row, k0+khalf+1]
    b.x = wrow[k0 + khalf];                 // B[k, ncol] = W[ncol, k]
    b.y = wrow[k0 + khalf + 1];
    // D = A(16x4) * B(4x16) + C, fp32 in / fp32 accumulate
    acc = __builtin_amdgcn_wmma_f32_16x16x4_f32(
        /*neg_a=*/false, a, /*neg_b=*/false, b,
        /*c_mod=*/(short)0, acc, /*reuse_a=*/false, /*reuse_b=*/false);
  }

  // D layout: lanes 0-15 hold M=0..7 in acc[0..7] -> rows 0,1 = acc[0],acc[1].
  __shared__ float red[4][16][2];
  if (lane < 16) {
    red[wave][lane][0] = acc[0];
    red[wave][lane][1] = acc[1];
  }
  __syncthreads();
  if (threadIdx.x < 16) {
    const int t = threadIdx.x;
    const int n = n0 + t;
    const float s0 = red[0][t][0] + red[1][t][0] + red[2][t][0] + red[3][t][0];
    const float s1 = red[0][t][1] + red[1][t][1] + red[2][t][1] + red[3][t][1];
    const float bn = bias[n];
    O[n]         = s0 + bn;
    O[D_DIM + n] = s1 + bn;
  }
}

// out[b, s, :] = o1[b, :]  -- 16.8 MB streaming broadcast, float4 NT stores.
// o1 is 8 KB -> L2 resident; stores are the only HBM traffic.
__global__ __launch_bounds__(256) void bcast_rows(
    const v4f* __restrict__ o1,    // [2, 256] x float4
    v4f* __restrict__ out) {       // [2, 2048, 256] x float4
  const unsigned i = blockIdx.x * 256u + threadIdx.x;  // 0 .. 2*2048*256-1
  const unsigned d4 = i & 255u;
  const unsigned b  = (i >> 8) >> 11;                  // i / (256*2048)
  const v4f v = o1[(b << 8) | d4];
  __builtin_nontemporal_store(v, &out[i]);
}

extern "C" void kernel_launch(void* const* d_in, const int* in_sizes, int n_in,
                              void* d_out, int out_size, void* d_ws, size_t ws_size,
                              hipStream_t stream) {
  // setup_inputs order:
  // 0 hidden_states, 1 condition, 2 Wq, 3 bq, 4 Wk, 5 bk, 6 Wv, 7 bv, 8 Wo, 9 bo
  const float* cond = (const float*)d_in[1];
  const float* Wv   = (const float*)d_in[6];
  const float* bv   = (const float*)d_in[7];
  const float* Wo   = (const float*)d_in[8];
  const float* bo   = (const float*)d_in[9];

  float* v1 = (float*)d_ws;            // [2, 1024]
  float* o1 = v1 + B_DIM * D_DIM;      // [2, 1024]

  // v1 = condition @ Wv^T + bv
  smallgemm_wmma_f32<<<64, 128, 0, stream>>>(cond, Wv, bv, v1);
  // o1 = v1 @ Wo^T + bo
  smallgemm_wmma_f32<<<64, 128, 0, stream>>>(v1, Wo, bo, o1);
  // out[b,s,:] = o1[b,:]
  const unsigned total4 = B_DIM * S_DIM * (D_DIM / 4);   // 1,048,576
  bcast_rows<<<total4 / 256, 256, 0, stream>>>((const v4f*)o1,
                                               (v4f*)d_out);
}